// GraphSAGE_4947802325460
// MI455X (gfx1250) — compile-verified
//
#include <hip/hip_runtime.h>
#include <hip/hip_bf16.h>

#define N_NODES 100000
#define D 128

typedef float v2f __attribute__((ext_vector_type(2)));
typedef float v8f __attribute__((ext_vector_type(8)));

// ---------------------------------------------------------------- utilities

__global__ void zero_kernel(float* __restrict__ p, long n) {
    long i = ((long)blockIdx.x * blockDim.x + threadIdx.x) * 4;
    if (i < n) *(float4*)(p + i) = make_float4(0.f, 0.f, 0.f, 0.f);
}

__global__ void degree_kernel(const int* __restrict__ dst, float* __restrict__ deg, int nE) {
    int e = blockIdx.x * blockDim.x + threadIdx.x;
    if (e < nE) unsafeAtomicAdd(&deg[dst[e]], 1.0f);
}

__global__ void invdeg_kernel(float* __restrict__ deg, int n) {
    int i = blockIdx.x * blockDim.x + threadIdx.x;
    if (i < n) {
        float d = deg[i];
        deg[i] = 1.0f / fmaxf(d, 1.0f);
    }
}

// ------------------------------------------------------- edge aggregation
// One wave32 per edge: each lane moves 4 of the 128 floats of the row.
// h row = 512B contiguous -> one b128 load per lane, 4 f32 atomics per lane.

__global__ __launch_bounds__(256)
void agg_kernel(const float* __restrict__ h, const int* __restrict__ src,
                const int* __restrict__ dst, float* __restrict__ neigh, int nE) {
    int e = (int)(((long)blockIdx.x * blockDim.x + threadIdx.x) >> 5);
    if (e >= nE) return;
    int lane = threadIdx.x & 31;
    int s = src[e];
    int t = dst[e];
    const float4 v = *(const float4*)(h + (size_t)s * D + lane * 4);
    float* b = neigh + (size_t)t * D + lane * 4;
    unsafeAtomicAdd(b + 0, v.x);
    unsafeAtomicAdd(b + 1, v.y);
    unsafeAtomicAdd(b + 2, v.z);
    unsafeAtomicAdd(b + 3, v.w);
}

// ------------------------------------------------------------ WMMA GEMM
// out[m,:] = h[m,:] @ Wself + (invdeg[m]*neigh[m,:]) @ Wneigh + bias  (+ReLU)
// Block: 256 threads = 8 waves -> 128 rows x 128 cols tile.
// Each wave: 16-row strip, 8 N-tiles of 16, fp32 WMMA 16x16x4.
// Two phases (self, neigh) sharing one 64KB LDS weight buffer.
//
// f32 WMMA layouts (ISA 7.12.2):
//   A 16x4 : lane L holds A[M=L%16, 2*(L/16)+j] in vgpr j  -> v2f {a[k],a[k+1]}
//   B 4x16 : lane L holds B[2*(L/16)+j, N=L%16] in vgpr j  -> v2f {b[k][n],b[k+1][n]}
//   C 16x16: lane L vgpr v holds C[M=v+8*(L/16), N=L%16]

__global__ __launch_bounds__(256)
void sage_gemm_kernel(const float* __restrict__ hs, const float* __restrict__ neigh,
                      const float* __restrict__ invdeg,
                      const float* __restrict__ Wself, const float* __restrict__ Wneigh,
                      const float* __restrict__ bias, float* __restrict__ out, int relu) {
    __shared__ float ldsW[D * D];  // 64 KB: one [K=128][N=128] weight matrix

    const int tid  = threadIdx.x;
    const int wave = tid >> 5;
    const int lane = tid & 31;
    const int half = lane >> 4;  // 0 or 1
    const int l16  = lane & 15;
    const int rowBase = blockIdx.x * 128 + wave * 16;
    int rowA = rowBase + l16;
    if (rowA >= N_NODES) rowA = N_NODES - 1;  // clamp for safe loads; stores guarded
    const int klane = 2 * half;

    v8f acc[8] = {};

    // ---------------- phase 0: self-term ----------------
    for (int i = tid; i < D * D / 4; i += 256)
        ((float4*)ldsW)[i] = ((const float4*)Wself)[i];
    __syncthreads();

    {
        const float* aRow = hs + (size_t)rowA * D;
#pragma unroll 4
        for (int k0 = 0; k0 < D; k0 += 4) {
            const int kk = k0 + klane;
            const float2 af = *(const float2*)(aRow + kk);
            v2f a; a.x = af.x; a.y = af.y;
#pragma unroll
            for (int nt = 0; nt < 8; ++nt) {
                v2f b;
                b.x = ldsW[kk * D + nt * 16 + l16];
                b.y = ldsW[(kk + 1) * D + nt * 16 + l16];
                acc[nt] = __builtin_amdgcn_wmma_f32_16x16x4_f32(
                    false, a, false, b, (short)0, acc[nt], false, false);
            }
        }
    }
    __syncthreads();

    // ---------------- phase 1: neighbor-term ----------------
    for (int i = tid; i < D * D / 4; i += 256)
        ((float4*)ldsW)[i] = ((const float4*)Wneigh)[i];
    __syncthreads();

    {
        const float s = invdeg[rowA];
        const float* nRow = neigh + (size_t)rowA * D;
#pragma unroll 4
        for (int k0 = 0; k0 < D; k0 += 4) {
            const int kk = k0 + klane;
            const float2 af = *(const float2*)(nRow + kk);
            v2f a; a.x = af.x * s; a.y = af.y * s;
#pragma unroll
            for (int nt = 0; nt < 8; ++nt) {
                v2f b;
                b.x = ldsW[kk * D + nt * 16 + l16];
                b.y = ldsW[(kk + 1) * D + nt * 16 + l16];
                acc[nt] = __builtin_amdgcn_wmma_f32_16x16x4_f32(
                    false, a, false, b, (short)0, acc[nt], false, false);
            }
        }
    }

    // ---------------- epilogue: bias, relu, store ----------------
#pragma unroll
    for (int nt = 0; nt < 8; ++nt) {
        const float bv = bias[nt * 16 + l16];
#pragma unroll
        for (int v = 0; v < 8; ++v) {
            const int row = rowBase + v + 8 * half;
            if (row < N_NODES) {
                float x = acc[nt][v] + bv;
                if (relu) x = fmaxf(x, 0.0f);
                out[(size_t)row * D + nt * 16 + l16] = x;
            }
        }
    }
}

// ---------------------------------------------------------------- launch

extern "C" void kernel_launch(void* const* d_in, const int* in_sizes, int n_in,
                              void* d_out, int out_size, void* d_ws, size_t ws_size,
                              hipStream_t stream) {
    const float* h0     = (const float*)d_in[0];
    const int*   src    = (const int*)d_in[1];   // harness delivers integer inputs as int32
    const int*   dst    = (const int*)d_in[2];
    const float* Wself  = (const float*)d_in[3];
    const float* Wneigh = (const float*)d_in[4];
    const float* bias   = (const float*)d_in[5];
    float*       out    = (float*)d_out;
    const int    nE     = in_sizes[1];

    // workspace layout: invdeg (N f32) | hA (N*D f32) | neigh (N*D f32)  ~103 MB
    char* ws = (char*)d_ws;
    float* invdeg = (float*)ws;
    float* hA     = (float*)(ws + 512 * 1024);
    float* neigh  = (float*)(ws + 512 * 1024 + (size_t)N_NODES * D * sizeof(float));

    const long nFeat = (long)N_NODES * D;
    const int  zeroFeatBlocks = (int)((nFeat / 4 + 255) / 256);
    const int  zeroDegBlocks  = (N_NODES / 4 + 255) / 256;
    const int  aggBlocks      = (nE + 7) / 8;       // 8 waves (edges) per block
    const int  gemmBlocks     = (N_NODES + 127) / 128;

    // in-degree -> 1/max(deg,1), in place
    zero_kernel<<<zeroDegBlocks, 256, 0, stream>>>(invdeg, (long)N_NODES);
    degree_kernel<<<(nE + 255) / 256, 256, 0, stream>>>(dst, invdeg, nE);
    invdeg_kernel<<<(N_NODES + 255) / 256, 256, 0, stream>>>(invdeg, N_NODES);

    // layer 0: h0 -> hA (relu)
    zero_kernel<<<zeroFeatBlocks, 256, 0, stream>>>(neigh, nFeat);
    agg_kernel<<<aggBlocks, 256, 0, stream>>>(h0, src, dst, neigh, nE);
    sage_gemm_kernel<<<gemmBlocks, 256, 0, stream>>>(h0, neigh, invdeg,
                                                     Wself, Wneigh, bias, hA, 1);

    // layer 1: hA -> out (relu)
    zero_kernel<<<zeroFeatBlocks, 256, 0, stream>>>(neigh, nFeat);
    agg_kernel<<<aggBlocks, 256, 0, stream>>>(hA, src, dst, neigh, nE);
    sage_gemm_kernel<<<gemmBlocks, 256, 0, stream>>>(hA, neigh, invdeg,
                                                     Wself + D * D, Wneigh + D * D,
                                                     bias + D, out, 1);

    // layer 2: out -> out (no relu; row-wise in-place is safe: GEMM row m reads
    // only row m of its input after aggregation completed in the prior kernel)
    zero_kernel<<<zeroFeatBlocks, 256, 0, stream>>>(neigh, nFeat);
    agg_kernel<<<aggBlocks, 256, 0, stream>>>(out, src, dst, neigh, nE);
    sage_gemm_kernel<<<gemmBlocks, 256, 0, stream>>>(out, neigh, invdeg,
                                                     Wself + 2 * D * D, Wneigh + 2 * D * D,
                                                     bias + 2 * D, out, 0);
}